// VEMGNN_71880572665977
// MI455X (gfx1250) — compile-verified
//
#include <hip/hip_runtime.h>
#include <hip/hip_bf16.h>

// ---------------------------------------------------------------------------
// GNN inference: encoder -> 2x (GraphConv linear + masked-max aggr + LN) -> decoder
// N=2048 nodes, F=16, D=64, L=2.
//  * All dense linears: V_WMMA_F32_16X16X4_F32 (exact f32, FLOPs negligible).
//  * Aggregation (the bandwidth stage, ~34MB adjacency): double-buffered
//    GLOBAL_LOAD_ASYNC_TO_LDS_B128 + S_WAIT_ASYNCCNT pipeline.
// ---------------------------------------------------------------------------

#define NN 2048
#define FF 16
#define DD 64
#define DH 32   // D/2

typedef float v2f __attribute__((ext_vector_type(2)));
typedef float v8f __attribute__((ext_vector_type(8)));

// ---- f32 WMMA fragment loaders (CDNA5 16x16x4 layouts, wave32) -------------
// A (16x4, MxK): lanes 0-15 -> M=lane, {v0=K0, v1=K1}; lanes 16-31 -> {K2, K3}
__device__ __forceinline__ v2f ldA(const float* A, int lda, int m0, int k0, int lane) {
    int row = m0 + (lane & 15);
    int kk  = k0 + ((lane >> 4) << 1);
    v2f a;
    a.x = A[row * lda + kk];
    a.y = A[row * lda + kk + 1];
    return a;
}
// B (4x16, KxN): lanes 0-15 -> N=lane, {v0=K0, v1=K1}; lanes 16-31 -> {K2, K3}
__device__ __forceinline__ v2f ldB(const float* B, int ldb, int k0, int n0, int lane) {
    int col = n0 + (lane & 15);
    int kk  = k0 + ((lane >> 4) << 1);
    v2f b;
    b.x = B[kk * ldb + col];
    b.y = B[(kk + 1) * ldb + col];
    return b;
}

#define WMMA_F32X4(acc, a, b) \
    acc = __builtin_amdgcn_wmma_f32_16x16x4_f32(false, (a), false, (b), (short)0, (acc), false, false)

// ---- CDNA5 async global->LDS copy (16B per lane, tracked by ASYNCcnt) ------
// VDST = per-lane LDS byte offset VGPR, VADDR = per-lane 64-bit global address.
__device__ __forceinline__ void async_copy_b128(unsigned lds_off, const void* gaddr) {
    asm volatile("global_load_async_to_lds_b128 %0, %1, off"
                 :: "v"(lds_off), "v"(gaddr)
                 : "memory");
}
#define WAIT_ASYNC_LE6() asm volatile("s_wait_asynccnt 0x6" ::: "memory")
#define WAIT_ASYNC_LE0() asm volatile("s_wait_asynccnt 0x0" ::: "memory")

// ---------------------------------------------------------------------------
// Encoder: h = relu(X @ w1 + b1) @ w2 + b2      [2048,16]->[2048,32]->[2048,64]
// One wave per 16-row tile.
// ---------------------------------------------------------------------------
__global__ __launch_bounds__(32)
void gnn_encoder_kernel(const float* __restrict__ X,
                        const float* __restrict__ w1, const float* __restrict__ b1,
                        const float* __restrict__ w2, const float* __restrict__ b2,
                        float* __restrict__ h) {
    __shared__ float t[16 * DH];             // relu(X@w1+b1) tile
    const int lane = threadIdx.x;
    const int m0   = blockIdx.x * 16;
    const int col16 = lane & 15;
    const int rbase = (lane >> 4) * 8;

    // stage 1: [16x16] x [16x32]
    for (int nt = 0; nt < 2; ++nt) {
        const int n0 = nt * 16;
        v8f acc = {};
        for (int k0 = 0; k0 < FF; k0 += 4) {
            v2f a = ldA(X, FF, m0, k0, lane);
            v2f b = ldB(w1, DH, k0, n0, lane);
            WMMA_F32X4(acc, a, b);
        }
        const int c = n0 + col16;
        const float bias = b1[c];
        #pragma unroll
        for (int r = 0; r < 8; ++r)
            t[(rbase + r) * DH + c] = fmaxf(acc[r] + bias, 0.0f);
    }
    __syncthreads();

    // stage 2: [16x32] x [32x64]
    for (int nt = 0; nt < 4; ++nt) {
        const int n0 = nt * 16;
        v8f acc = {};
        for (int k0 = 0; k0 < DH; k0 += 4) {
            v2f a = ldA(t, DH, 0, k0, lane);
            v2f b = ldB(w2, DD, k0, n0, lane);
            WMMA_F32X4(acc, a, b);
        }
        const int c = n0 + col16;
        const float bias = b2[c];
        #pragma unroll
        for (int r = 0; r < 8; ++r)
            h[(m0 + rbase + r) * DD + c] = acc[r] + bias;
    }
}

// ---------------------------------------------------------------------------
// GraphConv linear: xw = h @ w + b              [2048,64] x [64,64]
// One wave per 16x16 output tile; grid (128, 4).
// ---------------------------------------------------------------------------
__global__ __launch_bounds__(32)
void gnn_gclin_kernel(const float* __restrict__ h,
                      const float* __restrict__ w, const float* __restrict__ b,
                      float* __restrict__ xw) {
    const int lane = threadIdx.x;
    const int m0 = blockIdx.x * 16;
    const int n0 = blockIdx.y * 16;

    v8f acc = {};
    for (int k0 = 0; k0 < DD; k0 += 4) {
        v2f a  = ldA(h, DD, m0, k0, lane);
        v2f bf = ldB(w, DD, k0, n0, lane);
        WMMA_F32X4(acc, a, bf);
    }
    const int c = n0 + (lane & 15);
    const int rbase = (lane >> 4) * 8;
    const float bias = b[c];
    #pragma unroll
    for (int r = 0; r < 8; ++r)
        xw[(m0 + rbase + r) * DD + c] = acc[r] + bias;
}

// ---------------------------------------------------------------------------
// Masked max-aggregation + residual + LayerNorm (+optional ReLU).
// Block = 256 threads = 32 targets x 8 feature groups (8 feats each).
// Double-buffered async staging: per chunk, each wave issues exactly 6
// global_load_async_to_lds_b128 (4 for xw 16KB, 2 for adj 8KB). Async loads
// retire in order, so s_wait_asynccnt 6 after issuing chunk k+1 guarantees
// chunk k is resident; a barrier then publishes it block-wide.
// ---------------------------------------------------------------------------
#define JCHUNK 64
__global__ __launch_bounds__(256)
void gnn_aggregate_kernel(const float* __restrict__ xw,
                          const float* __restrict__ hin,
                          const int*   __restrict__ adj,
                          const float* __restrict__ ln_g,
                          const float* __restrict__ ln_b,
                          float* __restrict__ hout,
                          int do_relu) {
    __shared__ __align__(16) float s_xw[2][JCHUNK * DD];   // 2 x 16 KB
    __shared__ __align__(16) int   s_adj[2][JCHUNK * 32];  // 2 x 8 KB
    __shared__ float s_sum[8 * 32];
    __shared__ float s_sq[8 * 32];

    const int tid = threadIdx.x;
    const int tx = tid & 31;                 // target within tile
    const int ty = tid >> 5;                 // feature group
    const int ibase = blockIdx.x * 32;
    const int i = ibase + tx;

    // issue one chunk's async copies (6 instructions per wave)
    auto issue_chunk = [&](int bsel, int j0) {
        const float* gxw = xw + (size_t)j0 * DD;
        #pragma unroll
        for (int r = 0; r < 4; ++r) {                       // 1024 x 16B units
            const int u = tid + r * 256;
            async_copy_b128((unsigned)(uintptr_t)&s_xw[bsel][u * 4], gxw + u * 4);
        }
        #pragma unroll
        for (int r = 0; r < 2; ++r) {                       // 512 x 16B units
            const int u  = tid + r * 256;
            const int jj = u >> 3;
            const int q  = (u & 7) * 4;                     // int column within tile
            async_copy_b128((unsigned)(uintptr_t)&s_adj[bsel][jj * 32 + q],
                            adj + (size_t)(j0 + jj) * NN + ibase + q);
        }
    };

    float m[8];
    #pragma unroll
    for (int q = 0; q < 8; ++q) m[q] = -3.402823466e38f;

    issue_chunk(0, 0);
    int buf = 0;
    for (int j0 = 0; j0 < NN; j0 += JCHUNK) {
        if (j0 + JCHUNK < NN) {
            issue_chunk(buf ^ 1, j0 + JCHUNK);
            WAIT_ASYNC_LE6();                // oldest 6 (current chunk) retired
        } else {
            WAIT_ASYNC_LE0();
        }
        __syncthreads();                     // publish chunk block-wide

        for (int jj = 0; jj < JCHUNK; ++jj) {
            const int j = j0 + jj;
            const bool msk = (s_adj[buf][jj * 32 + tx] > 0) | (j == i);  // self loop
            const float* row = &s_xw[buf][jj * DD + ty * 8];
            #pragma unroll
            for (int q = 0; q < 8; ++q)
                m[q] = msk ? fmaxf(m[q], row[q]) : m[q];
        }
        __syncthreads();                     // all reads done before buf is refilled
        buf ^= 1;
    }

    // residual + LayerNorm partials
    float v[8], sum = 0.0f, sq = 0.0f;
    #pragma unroll
    for (int q = 0; q < 8; ++q) {
        v[q] = m[q] + hin[i * DD + ty * 8 + q];
        sum += v[q];
        sq  += v[q] * v[q];
    }
    s_sum[ty * 32 + tx] = sum;
    s_sq [ty * 32 + tx] = sq;
    __syncthreads();

    float tot = 0.0f, totsq = 0.0f;
    #pragma unroll
    for (int g = 0; g < 8; ++g) {
        tot   += s_sum[g * 32 + tx];
        totsq += s_sq [g * 32 + tx];
    }
    const float mu   = tot * (1.0f / 64.0f);
    const float var  = totsq * (1.0f / 64.0f) - mu * mu;
    const float rstd = rsqrtf(var + 1e-5f);

    #pragma unroll
    for (int q = 0; q < 8; ++q) {
        const int d = ty * 8 + q;
        float o = (v[q] - mu) * rstd * ln_g[d] + ln_b[d];
        if (do_relu) o = fmaxf(o, 0.0f);
        hout[i * DD + d] = o;
    }
}

// ---------------------------------------------------------------------------
// Decoder: out = relu(relu(h @ w1 + b1) @ w2 + b2)   [2048,64]->[2048,32]->[2048,1]
// One wave per 16-row tile; first layer via WMMA, final matvec scalar.
// ---------------------------------------------------------------------------
__global__ __launch_bounds__(32)
void gnn_decoder_kernel(const float* __restrict__ h,
                        const float* __restrict__ w1, const float* __restrict__ b1,
                        const float* __restrict__ w2, const float* __restrict__ b2,
                        float* __restrict__ out) {
    __shared__ float t[16 * DH];
    const int lane = threadIdx.x;
    const int m0 = blockIdx.x * 16;

    for (int nt = 0; nt < 2; ++nt) {
        const int n0 = nt * 16;
        v8f acc = {};
        for (int k0 = 0; k0 < DD; k0 += 4) {
            v2f a = ldA(h, DD, m0, k0, lane);
            v2f b = ldB(w1, DH, k0, n0, lane);
            WMMA_F32X4(acc, a, b);
        }
        const int c = n0 + (lane & 15);
        const int rbase = (lane >> 4) * 8;
        const float bias = b1[c];
        #pragma unroll
        for (int r = 0; r < 8; ++r)
            t[(rbase + r) * DH + c] = fmaxf(acc[r] + bias, 0.0f);
    }
    __syncthreads();

    if (lane < 16) {
        float s = b2[0];
        #pragma unroll
        for (int k = 0; k < DH; ++k)
            s += t[lane * DH + k] * w2[k];
        out[m0 + lane] = fmaxf(s, 0.0f);
    }
}

// ---------------------------------------------------------------------------
extern "C" void kernel_launch(void* const* d_in, const int* in_sizes, int n_in,
                              void* d_out, int out_size, void* d_ws, size_t ws_size,
                              hipStream_t stream) {
    (void)in_sizes; (void)n_in; (void)out_size; (void)ws_size;

    const float* X      = (const float*)d_in[0];
    const int*   A      = (const int*)  d_in[1];
    const float* enc_w1 = (const float*)d_in[2];
    const float* enc_b1 = (const float*)d_in[3];
    const float* enc_w2 = (const float*)d_in[4];
    const float* enc_b2 = (const float*)d_in[5];
    const float* gc_w   = (const float*)d_in[6];   // [L, D, D]
    const float* gc_b   = (const float*)d_in[7];   // [L, D]
    const float* ln_g   = (const float*)d_in[8];   // [L, D]
    const float* ln_b   = (const float*)d_in[9];   // [L, D]
    const float* dec_w1 = (const float*)d_in[10];
    const float* dec_b1 = (const float*)d_in[11];
    const float* dec_w2 = (const float*)d_in[12];
    const float* dec_b2 = (const float*)d_in[13];

    float* h0 = (float*)d_ws;                      // [2048, 64]
    float* h1 = h0 + NN * DD;                      // [2048, 64]
    float* xw = h1 + NN * DD;                      // [2048, 64]

    gnn_encoder_kernel<<<NN / 16, 32, 0, stream>>>(X, enc_w1, enc_b1, enc_w2, enc_b2, h0);

    float* hin = h0;
    float* hout = h1;
    for (int l = 0; l < 2; ++l) {
        gnn_gclin_kernel<<<dim3(NN / 16, DD / 16), 32, 0, stream>>>(
            hin, gc_w + l * DD * DD, gc_b + l * DD, xw);
        gnn_aggregate_kernel<<<NN / 32, 256, 0, stream>>>(
            xw, hin, A, ln_g + l * DD, ln_b + l * DD, hout, (l < 1) ? 1 : 0);
        float* tmp = hin; hin = hout; hout = tmp;
    }

    gnn_decoder_kernel<<<NN / 16, 32, 0, stream>>>(hin, dec_w1, dec_b1, dec_w2, dec_b2,
                                                   (float*)d_out);
}